// E63aNonlinearDeltaCell_4887672783118
// MI455X (gfx1250) — compile-verified
//
#include <hip/hip_runtime.h>
#include <hip/hip_bf16.h>

#define T_DIM 1024
#define B_DIM 32
#define D_DIM 1024
#define M_DIM (T_DIM * B_DIM)   // 32768 flattened (t,b) rows

#define SCAN_NWG 32             // persistent workgroups in scan kernel
#define SCAN_NB  32             // N columns owned per scan workgroup
#define WSTRIDE  1028           // padded row stride (dwords) -> conflict-free b64 reads
#define H_LDS_OFF_DW (SCAN_NB * WSTRIDE)          // staged h starts after W slice
#define H_LDS_OFF_BYTES (H_LDS_OFF_DW * 4)        // 131584
#define SCAN_LDS_BYTES ((SCAN_NB * WSTRIDE + B_DIM * WSTRIDE) * 4)  // ~263 KB

typedef __attribute__((ext_vector_type(2))) float v2f;
typedef __attribute__((ext_vector_type(8))) float v8f;
typedef unsigned int u32x4 __attribute__((ext_vector_type(4)));
typedef int i32x4 __attribute__((ext_vector_type(4)));
typedef int i32x8 __attribute__((ext_vector_type(8)));

#if __has_include(<hip/amd_detail/amd_gfx1250_TDM.h>)
#define TDM_6ARG 1
#else
#define TDM_6ARG 0
#endif

__device__ __forceinline__ v8f wmma_f32x4(v2f a, v2f b, v8f c) {
  // D = A(16x4) * B(4x16) + C(16x16), fp32 WMMA
  return __builtin_amdgcn_wmma_f32_16x16x4_f32(false, a, false, b, (short)0, c,
                                               false, false);
}

// ---------------------------------------------------------------------------
// TDM: DMA one full h_prev tile [32 x 1024] f32 from global into LDS at
// H_LDS_OFF, with 1 dword of padding per 256 dwords (row stride -> 1028 dw).
// D# built per CDNA5 ISA ch.8 bitfields.
// ---------------------------------------------------------------------------
#if __has_builtin(__builtin_amdgcn_tensor_load_to_lds)
#define HAVE_TDM 1
__device__ __forceinline__ void tdm_load_h(const float* gsrc) {
  const unsigned long long ga = (unsigned long long)(uintptr_t)gsrc;
  // group0: count=1 | lds_addr | global_addr[31:0] | global_addr[56:32], type=2
  u32x4 g0 = {1u, (unsigned)H_LDS_OFF_BYTES, (unsigned)ga,
              (unsigned)((ga >> 32) & 0x01FFFFFFu) | 0x80000000u};
  // group1: data_size=4B(2<<16) | pad_enable(1<<20) | pad_interval=256dw(7<<22)
  //         | pad_amount=1dw(0<<25); tensor_dim0=1024, tensor_dim1=32,
  //         tile_dim0=1024, tile_dim1=32, tensor_dim0_stride=1024
  i32x8 g1 = {(int)0x01D20000u,       // mask=0, dsize=2, pad cfg
              (int)(1024u << 16),     // abar=0 | tensor_dim0[15:0]
              (int)(32u << 16),       // tensor_dim0[31:16]=0 | tensor_dim1[15:0]
              (int)(1024u << 16),     // tensor_dim1[31:16]=0 | tile_dim0
              32,                     // tile_dim1=32, tile_dim2=0
              1024,                   // tensor_dim0_stride[31:0]
              0, 0};
  i32x4 z4 = {0, 0, 0, 0};
#if TDM_6ARG
  i32x8 z8 = {0, 0, 0, 0, 0, 0, 0, 0};
  __builtin_amdgcn_tensor_load_to_lds(g0, g1, z4, z4, z8, 0);
#else
  __builtin_amdgcn_tensor_load_to_lds(g0, g1, z4, z4, 0);
#endif
  __builtin_amdgcn_s_wait_tensorcnt(0);
}
#else
#define HAVE_TDM 0
#endif

// ---------------------------------------------------------------------------
// Kernel A: alpha_all = sigmoid(x @ W_alpha^T + b_alpha), wx_all = x @ W_x^T + b
// One wave computes a 16(M) x 64(N) tile of BOTH outputs (shares A loads).
// ---------------------------------------------------------------------------
__global__ __launch_bounds__(256) void precompute_kernel(
    const float* __restrict__ x,  const float* __restrict__ Wa,
    const float* __restrict__ ba, const float* __restrict__ Wx,
    const float* __restrict__ bb,
    float* __restrict__ alpha_ws, float* __restrict__ wx_ws) {
  const int wave  = threadIdx.x >> 5;
  const int lane  = threadIdx.x & 31;
  const int mtile = blockIdx.x >> 1;
  const int nblk  = (blockIdx.x & 1) * 512 + wave * 64;

  const int lrow  = lane & 15;
  const int khalf = (lane >> 4) * 2;

  v8f accA[4], accX[4];
#pragma unroll
  for (int j = 0; j < 4; ++j) {
    accA[j] = (v8f){0.f, 0.f, 0.f, 0.f, 0.f, 0.f, 0.f, 0.f};
    accX[j] = (v8f){0.f, 0.f, 0.f, 0.f, 0.f, 0.f, 0.f, 0.f};
  }

  const float* xrow = x + (size_t)(mtile * 16 + lrow) * D_DIM + khalf;
  const float* waRow[4];
  const float* wxRow[4];
#pragma unroll
  for (int j = 0; j < 4; ++j) {
    const int n = nblk + j * 16 + lrow;
    waRow[j] = Wa + (size_t)n * D_DIM + khalf;
    wxRow[j] = Wx + (size_t)n * D_DIM + khalf;
  }

#pragma unroll 4
  for (int k0 = 0; k0 < D_DIM; k0 += 4) {
    const v2f a = *(const v2f*)(xrow + k0);
#pragma unroll
    for (int j = 0; j < 4; ++j) {
      const v2f bA = *(const v2f*)(waRow[j] + k0);
      accA[j] = wmma_f32x4(a, bA, accA[j]);
      const v2f bX = *(const v2f*)(wxRow[j] + k0);
      accX[j] = wmma_f32x4(a, bX, accX[j]);
    }
  }

#pragma unroll
  for (int j = 0; j < 4; ++j) {
    const int n = nblk + j * 16 + lrow;
    const float ban = ba[n];
    const float bn  = bb[n];
#pragma unroll
    for (int jj = 0; jj < 8; ++jj) {
      const int m = mtile * 16 + jj + (lane >> 4) * 8;
      const size_t idx = (size_t)m * D_DIM + n;
      const float za = accA[j][jj] + ban;
      alpha_ws[idx] = 1.f / (1.f + __expf(-za));
      wx_ws[idx]    = accX[j][jj] + bn;
    }
  }
}

// ---------------------------------------------------------------------------
// Grid-wide barrier for the persistent scan kernel
// ---------------------------------------------------------------------------
__device__ __forceinline__ void grid_barrier(unsigned* counter, unsigned target) {
  __threadfence();
  __syncthreads();
  if (threadIdx.x == 0) {
    __hip_atomic_fetch_add(counter, 1u, __ATOMIC_RELEASE, __HIP_MEMORY_SCOPE_AGENT);
    while (__hip_atomic_load(counter, __ATOMIC_ACQUIRE, __HIP_MEMORY_SCOPE_AGENT) <
           target) {
      __builtin_amdgcn_s_sleep(1);
    }
  }
  __syncthreads();
  __threadfence();
}

// ---------------------------------------------------------------------------
// Kernel B: persistent sequential scan.
// 32 WGs x 128 threads (4 waves). WG owns 32 N-columns; W_h slice lives in LDS
// for the whole scan; h_prev (32x1024) is TDM-staged into LDS each step with
// padding so both A and B feeds are conflict-free ds_load_b64.
// ---------------------------------------------------------------------------
__global__ __launch_bounds__(128) void scan_kernel(
    const float* __restrict__ h0, const float* __restrict__ Wh,
    const float* __restrict__ alpha_ws, const float* __restrict__ wx_ws,
    float* __restrict__ out, float* hout, unsigned* counter) {
  extern __shared__ float lds[];
  float* lds_w = lds;                 // [SCAN_NB][WSTRIDE]
  float* lds_h = lds + H_LDS_OFF_DW;  // [B_DIM rows, padded stride 1028 dw]

  const int tid  = threadIdx.x;
  const int wave = tid >> 5;
  const int lane = tid & 31;
  const int nblk = blockIdx.x * SCAN_NB;

  // Preload W_h slice into LDS (row-major, padded stride)
  for (int idx = tid; idx < SCAN_NB * D_DIM; idx += 128) {
    const int n = idx >> 10;
    const int k = idx & (D_DIM - 1);
    lds_w[n * WSTRIDE + k] = Wh[(size_t)(nblk + n) * D_DIM + k];
  }
  // h[0] = h0 (our column slice)
  for (int idx = tid; idx < B_DIM * SCAN_NB; idx += 128) {
    const int b = idx / SCAN_NB;
    const int n = nblk + (idx % SCAN_NB);
    hout[(size_t)b * D_DIM + n] = h0[(size_t)b * D_DIM + n];
  }
  grid_barrier(counter, SCAN_NWG);

  const int mtile = wave >> 1;                        // 0..1 (batch rows)
  const int ntile = wave & 1;                         // 0..1 (col tile)
  const int lrow  = lane & 15;
  const int khalf = (lane >> 4) * 2;
  const int arow  = mtile * 16 + lrow;                // batch index for A
  const int abase = arow * WSTRIDE + khalf;           // padded dword base in lds_h
  const float* ldsB = lds_w + (ntile * 16 + lrow) * WSTRIDE + khalf;
  const int ncol  = nblk + ntile * 16 + lrow;         // output column

  for (int t = 0; t < T_DIM; ++t) {
    const float* hprev = hout + (size_t)t * (B_DIM * D_DIM);
    float*       hnext = hout + (size_t)(t + 1) * (B_DIM * D_DIM);

    // Stage h_prev into LDS
#if HAVE_TDM
    if (wave == 0) tdm_load_h(hprev);
#else
    for (int idx = tid; idx < B_DIM * D_DIM; idx += 128) {
      const int r = idx >> 10, k = idx & (D_DIM - 1);
      lds_h[r * WSTRIDE + (k + (k >> 8))] = hprev[(size_t)r * D_DIM + k];
    }
#endif
    __syncthreads();

    v8f acc = (v8f){0.f, 0.f, 0.f, 0.f, 0.f, 0.f, 0.f, 0.f};
#pragma unroll
    for (int kc = 0; kc < 4; ++kc) {              // 4 chunks of 256 dwords
      const float* ap = lds_h + abase + kc * 257; // +1 pad dword per chunk
      const float* bp = ldsB + kc * 256;
#pragma unroll 16
      for (int k = 0; k < 256; k += 4) {
        const v2f a = *(const v2f*)(ap + k);      // h_prev (LDS, TDM-staged)
        const v2f b = *(const v2f*)(bp + k);      // W_h slice (LDS)
        acc = wmma_f32x4(a, b, acc);
      }
    }

    const size_t tbase = (size_t)t * (B_DIM * D_DIM);
#pragma unroll
    for (int jj = 0; jj < 8; ++jj) {
      const int m = mtile * 16 + jj + (lane >> 4) * 8;    // batch index
      const size_t idx = tbase + (size_t)m * D_DIM + ncol;
      const int hi = m * WSTRIDE + (ncol + (ncol >> 8));  // padded LDS index
      const float wx = wx_ws[idx];
      const float al = alpha_ws[idx];
      const float hp = lds_h[hi];
      const float v  = tanhf(acc[jj] + wx);
      const float hn = al * hp + (1.f - al) * v;
      out[idx] = hn * hn * (1.f / (1.f + __expf(-hn)));   // hn * silu(hn)
      hnext[(size_t)m * D_DIM + ncol] = hn;
    }
    __syncthreads();   // don't restage lds_h while others still read it
    grid_barrier(counter, (unsigned)(SCAN_NWG * (t + 2)));
  }
}

extern "C" void kernel_launch(void* const* d_in, const int* in_sizes, int n_in,
                              void* d_out, int out_size, void* d_ws, size_t ws_size,
                              hipStream_t stream) {
  const float* x  = (const float*)d_in[0];
  const float* h0 = (const float*)d_in[1];
  const float* Wa = (const float*)d_in[2];
  const float* ba = (const float*)d_in[3];
  const float* Wh = (const float*)d_in[4];
  const float* Wx = (const float*)d_in[5];
  const float* bb = (const float*)d_in[6];

  float* out  = (float*)d_out;                           // [T,B,D]
  float* hout = out + (size_t)T_DIM * B_DIM * D_DIM;     // [T+1,B,D]

  float*    alpha_ws = (float*)d_ws;
  float*    wx_ws    = alpha_ws + (size_t)T_DIM * B_DIM * D_DIM;
  unsigned* counter  = (unsigned*)(wx_ws + (size_t)T_DIM * B_DIM * D_DIM);

  hipMemsetAsync(counter, 0, 256, stream);               // reset grid barrier

  precompute_kernel<<<dim3((M_DIM / 16) * 2), dim3(256), 0, stream>>>(
      x, Wa, ba, Wx, bb, alpha_ws, wx_ws);

  (void)hipFuncSetAttribute((const void*)scan_kernel,
                            hipFuncAttributeMaxDynamicSharedMemorySize,
                            SCAN_LDS_BYTES);
  scan_kernel<<<dim3(SCAN_NWG), dim3(128), SCAN_LDS_BYTES, stream>>>(
      h0, Wh, alpha_ws, wx_ws, out, hout, counter);
}